// NEQUIP_11218454577154
// MI455X (gfx1250) — compile-verified
//
#include <hip/hip_runtime.h>

#define NN 50000
#define NE 800000

typedef float v2f __attribute__((ext_vector_type(2)));
typedef float v8f __attribute__((ext_vector_type(8)));

__device__ __forceinline__ v8f v8f_zero() {
  v8f z = {0.f,0.f,0.f,0.f,0.f,0.f,0.f,0.f};
  return z;
}

// D = A(16x4 f32) * B(4x16 f32) + C(16x16 f32), full-rate CDNA5 WMMA
__device__ __forceinline__ v8f wmma_k4(v2f a, v2f b, v8f c) {
  return __builtin_amdgcn_wmma_f32_16x16x4_f32(false, a, false, b, (short)0, c, false, false);
}

__device__ __forceinline__ float swishf(float x) { return x / (1.0f + __expf(-x)); }

#define SQRT3f      1.7320508075688772f
#define INV_SQRT3f  0.5773502691896258f
#define CG_A        0.31622776601683794f   /* 1/sqrt(10) */
#define CG_B        0.18257418583505536f   /* 1/sqrt(30) */
#define INV_SQRT_AVG 0.22360679774997896f  /* 1/sqrt(20) */

// ---------------------------------------------------------------------------
// Per-edge geometry: sh1 (3), sh2 (5), radial bessel*envelope (8)
// ---------------------------------------------------------------------------
__global__ void k_edge_geom(const float* __restrict__ nodes,
                            const int* __restrict__ snd,
                            const int* __restrict__ rcv,
                            float* __restrict__ radial,
                            float* __restrict__ sh1,
                            float* __restrict__ sh2) {
  int e = blockIdx.x * 256 + threadIdx.x;
  if (e >= NE) return;
  int s = snd[e], t = rcv[e];
  float dx = nodes[t*6+0] - nodes[s*6+0];
  float dy = nodes[t*6+1] - nodes[s*6+1];
  float dz = nodes[t*6+2] - nodes[s*6+2];
  float r  = sqrtf(dx*dx + dy*dy + dz*dz);
  float rs = (r == 0.0f) ? 1.0f : r;
  float inv = 1.0f / rs;
  float x = dx*inv, y = dy*inv, z = dz*inv;
  const float s15 = 3.872983346207417f;
  const float s5  = 2.23606797749979f;
  sh1[e*3+0] = SQRT3f*y; sh1[e*3+1] = SQRT3f*z; sh1[e*3+2] = SQRT3f*x;
  sh2[e*5+0] = s15*x*y;
  sh2[e*5+1] = s15*y*z;
  sh2[e*5+2] = 0.5f*s5*(3.0f*z*z - 1.0f);
  sh2[e*5+3] = s15*x*z;
  sh2[e*5+4] = 0.5f*s15*(x*x - y*y);
  float env = 0.0f;
  if (r < 1.0f) {
    float r2 = r*r, r3 = r2*r, r6 = r3*r3, r7 = r6*r, r8 = r7*r;
    env = 1.0f - 28.0f*r6 + 48.0f*r7 - 21.0f*r8;
  }
  float pre = (r == 0.0f) ? 0.0f : 1.4142135623730951f * inv * env;
  const float PI = 3.14159265358979323846f;
#pragma unroll
  for (int k = 1; k <= 8; ++k)
    radial[e*8 + (k-1)] = pre * __sinf(PI * (float)k * r);
}

// ---------------------------------------------------------------------------
// Per-node: h = v@up/sqrt(m), sc = v@sc/sqrt(m); zero aggregates + sumsq
// ---------------------------------------------------------------------------
__global__ void k_node_pre(const float* __restrict__ nodes,
                           const float* __restrict__ vbuf,
                           const float* __restrict__ upW,
                           const float* __restrict__ scW,
                           float* __restrict__ hbuf,
                           float* __restrict__ scbuf,
                           float* __restrict__ s_agg,
                           float* __restrict__ v_agg,
                           float* __restrict__ sumsq,
                           int m, int first) {
  int nd = blockIdx.x * 256 + threadIdx.x;
  if (nd >= NN) return;
  float vv[2][3];
#pragma unroll
  for (int i = 0; i < 3; ++i) { vv[0][i] = 0.f; vv[1][i] = 0.f; }
  if (first) {
#pragma unroll
    for (int i = 0; i < 3; ++i) vv[0][i] = nodes[nd*6 + 3 + i];
  } else {
    for (int mm = 0; mm < m; ++mm)
#pragma unroll
      for (int i = 0; i < 3; ++i) vv[mm][i] = vbuf[nd*6 + mm*3 + i];
  }
  float invsm = rsqrtf((float)m);
  for (int u = 0; u < m; ++u)
#pragma unroll
    for (int i = 0; i < 3; ++i) {
      float a = 0.f;
      for (int mm = 0; mm < m; ++mm) a += vv[mm][i] * upW[mm*m + u];
      hbuf[nd*6 + u*3 + i] = a * invsm;
    }
  for (int u = 0; u < 2; ++u)
#pragma unroll
    for (int i = 0; i < 3; ++i) {
      float a = 0.f;
      for (int mm = 0; mm < m; ++mm) a += vv[mm][i] * scW[mm*2 + u];
      scbuf[nd*6 + u*3 + i] = a * invsm;
    }
  s_agg[nd*2+0] = 0.f; s_agg[nd*2+1] = 0.f;
#pragma unroll
  for (int c = 0; c < 12; ++c) v_agg[nd*12 + c] = 0.f;
  if (nd == 0) { sumsq[0] = 0.f; sumsq[1] = 0.f; }
}

// ---------------------------------------------------------------------------
// Fused edge kernel: WMMA MLP (16 edges/wave) + gather + tensor product +
// scatter-add.  96 x v_wmma_f32_16x16x4_f32 per wave.
// ---------------------------------------------------------------------------
__launch_bounds__(256)
__global__ void k_edge_msg(const float* __restrict__ radial,
                           const float* __restrict__ sh1g,
                           const float* __restrict__ sh2g,
                           const float* __restrict__ gvec,
                           const float* __restrict__ hbuf,
                           const int* __restrict__ snd,
                           const int* __restrict__ rcv,
                           const float* __restrict__ w1,
                           const float* __restrict__ w2,
                           const float* __restrict__ w3,
                           float* __restrict__ s_agg,
                           float* __restrict__ v_agg,
                           int m) {
  __shared__ float w1s[16*72];    //  4608 B (bank stride 72)
  __shared__ float w2s[64*72];    // 18432 B
  __shared__ float w3s[64*24];    //  6144 B (zero-padded to 16 cols)
  __shared__ float gls[8];
  __shared__ float ht[8*16*68];   // 34816 B per-wave 16x64 activation tiles

  const int tid = threadIdx.x;
  for (int i = tid; i < 16*64; i += 256) w1s[(i>>6)*72 + (i&63)] = w1[i];
  for (int i = tid; i < 64*64; i += 256) w2s[(i>>6)*72 + (i&63)] = w2[i];
  const int w3n = 3*m;
  for (int i = tid; i < 64*16; i += 256) {
    int k = i >> 4, n = i & 15;
    w3s[k*24 + n] = (n < w3n) ? w3[k*w3n + n] : 0.0f;
  }
  if (tid < 8) gls[tid] = gvec[tid];
  __syncthreads();

  const int wv   = tid >> 5;
  const int lane = tid & 31;
  const int half = lane >> 4;
  const int l15  = lane & 15;           // M row (A) / N col (B,C)
  const int e0   = (blockIdx.x * 8 + wv) * 16;
  float* myht = &ht[wv * 16 * 68];

  // ---- GEMM1: x(16x16) @ w1(16x64), scale 1/4, swish ----
  v8f acc[4];
#pragma unroll
  for (int nt = 0; nt < 4; ++nt) acc[nt] = v8f_zero();
#pragma unroll
  for (int j = 0; j < 4; ++j) {
    int c = 4*j + 2*half;
    float a0, a1;
    if (c < 8) {
      int base = (e0 + l15)*8 + c;
      a0 = radial[base]; a1 = radial[base + 1];
    } else {
      a0 = gls[c - 8]; a1 = gls[c - 7];
    }
    v2f A = {a0, a1};
#pragma unroll
    for (int nt = 0; nt < 4; ++nt) {
      int n = nt*16 + l15;
      v2f B = { w1s[c*72 + n], w1s[(c+1)*72 + n] };
      acc[nt] = wmma_k4(A, B, acc[nt]);
    }
  }
#pragma unroll
  for (int r = 0; r < 8; ++r) {
    int row = r + 8*half;
#pragma unroll
    for (int nt = 0; nt < 4; ++nt)
      myht[row*68 + nt*16 + l15] = swishf(acc[nt][r] * 0.25f);
  }
  __syncthreads();

  // ---- GEMM2: h1(16x64) @ w2(64x64), scale 1/8, swish ----
#pragma unroll
  for (int nt = 0; nt < 4; ++nt) acc[nt] = v8f_zero();
#pragma unroll
  for (int j = 0; j < 16; ++j) {
    int c = 4*j + 2*half;
    v2f A = { myht[l15*68 + c], myht[l15*68 + c + 1] };
#pragma unroll
    for (int nt = 0; nt < 4; ++nt) {
      int n = nt*16 + l15;
      v2f B = { w2s[c*72 + n], w2s[(c+1)*72 + n] };
      acc[nt] = wmma_k4(A, B, acc[nt]);
    }
  }
  __syncthreads();
#pragma unroll
  for (int r = 0; r < 8; ++r) {
    int row = r + 8*half;
#pragma unroll
    for (int nt = 0; nt < 4; ++nt)
      myht[row*68 + nt*16 + l15] = swishf(acc[nt][r] * 0.125f);
  }
  __syncthreads();

  // ---- GEMM3: h2(16x64) @ w3(64x3m), scale 1/8 (no activation) ----
  v8f mcc = v8f_zero();
#pragma unroll
  for (int j = 0; j < 16; ++j) {
    int c = 4*j + 2*half;
    v2f A = { myht[l15*68 + c], myht[l15*68 + c + 1] };
    v2f B = { w3s[c*24 + l15], w3s[(c+1)*24 + l15] };
    mcc = wmma_k4(A, B, mcc);
  }
  __syncthreads();
  // reuse ht region as mix tile: [16 edges][9] per wave
  float* mymix = myht;
  if (l15 < 8) {
#pragma unroll
    for (int r = 0; r < 8; ++r)
      mymix[(r + 8*half)*9 + l15] = mcc[r] * 0.125f;
  }
  __syncthreads();

  // ---- Epilogue: gather msg, l=1x2->1 TP, scatter-add ----
  const int mm = half;          // multiplicity channel handled by this lane
  if (mm < m) {
    int e = e0 + l15;
    int s = snd[e];
    int t = rcv[e];
    float u0 = hbuf[s*6 + mm*3 + 0];
    float u1 = hbuf[s*6 + mm*3 + 1];
    float u2 = hbuf[s*6 + mm*3 + 2];
    float p0 = sh1g[e*3+0], p1 = sh1g[e*3+1], p2 = sh1g[e*3+2];
    float q0 = sh2g[e*5+0], q1 = sh2g[e*5+1], q2 = sh2g[e*5+2];
    float q3 = sh2g[e*5+3], q4 = sh2g[e*5+4];

    float stp = INV_SQRT3f * (u0*p0 + u1*p1 + u2*p2);
    // analytic CG121 (Gaunt 1x2->1, Frobenius-normalized)
    float t0 = SQRT3f * (CG_A*(u2*q0 + u1*q1 - u0*q4) - CG_B*u0*q2);
    float t1 = SQRT3f * (CG_A*u2*q3 + 2.0f*CG_B*u1*q2);
    float t2 = SQRT3f * (CG_A*(u0*q0 + u1*q3 + u2*q4) - CG_B*u2*q2);

    float mx_s   = mymix[l15*9 + mm];
    float mx_msg = mymix[l15*9 + m + mm];
    float mx_vtp = mymix[l15*9 + 2*m + mm];

    atomicAdd(&s_agg[t*2 + mm], stp * mx_s);
    atomicAdd(&v_agg[t*12 + mm*3 + 0], u0 * mx_msg);
    atomicAdd(&v_agg[t*12 + mm*3 + 1], u1 * mx_msg);
    atomicAdd(&v_agg[t*12 + mm*3 + 2], u2 * mx_msg);
    atomicAdd(&v_agg[t*12 + (m+mm)*3 + 0], t0 * mx_vtp);
    atomicAdd(&v_agg[t*12 + (m+mm)*3 + 1], t1 * mx_vtp);
    atomicAdd(&v_agg[t*12 + (m+mm)*3 + 2], t2 * mx_vtp);
  }
}

// ---------------------------------------------------------------------------
// Per-node post: downproject, gate, accumulate per-channel sum of squares
// ---------------------------------------------------------------------------
__global__ void k_node_post(const float* __restrict__ s_agg,
                            const float* __restrict__ v_agg,
                            const float* __restrict__ scbuf,
                            const float* __restrict__ dn_s,
                            const float* __restrict__ dn_v,
                            float* __restrict__ vtmp,
                            float* __restrict__ sumsq,
                            int m) {
  int nd = blockIdx.x * 256 + threadIdx.x;
  float loc0 = 0.f, loc1 = 0.f;
  if (nd < NN) {
    float invm  = rsqrtf((float)m);
    float inv2m = rsqrtf(2.0f*(float)m);
    float sa[2] = {0.f, 0.f};
    for (int mm = 0; mm < m; ++mm) sa[mm] = s_agg[nd*2 + mm] * INV_SQRT_AVG;
    float sw[2];
#pragma unroll
    for (int u = 0; u < 2; ++u) {
      float so = 0.f;
      for (int mm = 0; mm < m; ++mm) so += sa[mm] * dn_s[mm*2 + u];
      sw[u] = swishf(so * invm);
    }
#pragma unroll
    for (int u = 0; u < 2; ++u) {
#pragma unroll
      for (int i = 0; i < 3; ++i) {
        float a = 0.f;
        for (int c = 0; c < 2*m; ++c) a += v_agg[nd*12 + c*3 + i] * dn_v[c*2 + u];
        float vo = a * INV_SQRT_AVG * inv2m + scbuf[nd*6 + u*3 + i];
        float vn = vo * sw[u];
        vtmp[nd*6 + u*3 + i] = vn;
        if (u == 0) loc0 += vn*vn; else loc1 += vn*vn;
      }
    }
  }
#pragma unroll
  for (int off = 16; off > 0; off >>= 1) {
    loc0 += __shfl_down(loc0, off);
    loc1 += __shfl_down(loc1, off);
  }
  if ((threadIdx.x & 31) == 0) {
    atomicAdd(&sumsq[0], loc0);
    atomicAdd(&sumsq[1], loc1);
  }
}

__global__ void k_normalize(const float* __restrict__ vtmp,
                            const float* __restrict__ sumsq,
                            float* __restrict__ vbuf) {
  int nd = blockIdx.x * 256 + threadIdx.x;
  if (nd >= NN) return;
  float inv0 = rsqrtf(sumsq[0] / (3.0f*(float)NN) + 1e-5f);
  float inv1 = rsqrtf(sumsq[1] / (3.0f*(float)NN) + 1e-5f);
#pragma unroll
  for (int i = 0; i < 3; ++i) {
    vbuf[nd*6 + i]     = vtmp[nd*6 + i]     * inv0;
    vbuf[nd*6 + 3 + i] = vtmp[nd*6 + 3 + i] * inv1;
  }
}

__global__ void k_final(const float* __restrict__ nodes,
                        const float* __restrict__ vbuf,
                        float* __restrict__ out) {
  int nd = blockIdx.x * 256 + threadIdx.x;
  if (nd >= NN) return;
#pragma unroll
  for (int i = 0; i < 3; ++i) {
    out[nd*6 + i]     = nodes[nd*6 + i] + vbuf[nd*6 + i];
    out[nd*6 + 3 + i] = vbuf[nd*6 + 3 + i];
  }
}

// ---------------------------------------------------------------------------
extern "C" void kernel_launch(void* const* d_in, const int* in_sizes, int n_in,
                              void* d_out, int out_size, void* d_ws, size_t ws_size,
                              hipStream_t stream) {
  const float* nodes = (const float*)d_in[0];
  const float* gvec  = (const float*)d_in[1];
  // Param pytree order: jax sorted keys (dn_s,dn_v,sc,up,w1,w2,w3) vs dict
  // insertion order (sc,up,w1,w2,w3,dn_s,dn_v). Distinguish via in_sizes[3]:
  // layer0 dn_v has 4 elements (sorted) vs up has 1 element (insertion).
  bool sorted_keys = (n_in > 3) && (in_sizes[3] == 4);
  int i_sc[3], i_up[3], i_w1[3], i_w2[3], i_w3[3], i_dns[3], i_dnv[3];
  for (int l = 0; l < 3; ++l) {
    int b = 2 + l*7;
    if (sorted_keys) {
      i_dns[l]=b; i_dnv[l]=b+1; i_sc[l]=b+2; i_up[l]=b+3;
      i_w1[l]=b+4; i_w2[l]=b+5; i_w3[l]=b+6;
    } else {
      i_sc[l]=b; i_up[l]=b+1; i_w1[l]=b+2; i_w2[l]=b+3;
      i_w3[l]=b+4; i_dns[l]=b+5; i_dnv[l]=b+6;
    }
  }
  const int* snd = (const int*)d_in[n_in-2];
  const int* rcv = (const int*)d_in[n_in-1];

  float* ws = (float*)d_ws;
  size_t off = 0;
  float* radial = ws + off; off += (size_t)NE*8;
  float* sh1    = ws + off; off += (size_t)NE*3;
  float* sh2    = ws + off; off += (size_t)NE*5;
  float* vbuf   = ws + off; off += (size_t)NN*6;
  float* hbuf   = ws + off; off += (size_t)NN*6;
  float* scbuf  = ws + off; off += (size_t)NN*6;
  float* vtmp   = ws + off; off += (size_t)NN*6;
  float* s_agg  = ws + off; off += (size_t)NN*2;
  float* v_agg  = ws + off; off += (size_t)NN*12;
  float* sumsq  = ws + off; off += 2;
  (void)ws_size; (void)out_size; (void)in_sizes;

  const int nbN = (NN + 255)/256;
  k_edge_geom<<<(NE + 255)/256, 256, 0, stream>>>(nodes, snd, rcv, radial, sh1, sh2);

  for (int l = 0; l < 3; ++l) {
    int m = (l == 0) ? 1 : 2;
    k_node_pre<<<nbN, 256, 0, stream>>>(nodes, vbuf,
        (const float*)d_in[i_up[l]], (const float*)d_in[i_sc[l]],
        hbuf, scbuf, s_agg, v_agg, sumsq, m, (l == 0) ? 1 : 0);
    k_edge_msg<<<NE/128, 256, 0, stream>>>(radial, sh1, sh2, gvec, hbuf, snd, rcv,
        (const float*)d_in[i_w1[l]], (const float*)d_in[i_w2[l]],
        (const float*)d_in[i_w3[l]], s_agg, v_agg, m);
    k_node_post<<<nbN, 256, 0, stream>>>(s_agg, v_agg, scbuf,
        (const float*)d_in[i_dns[l]], (const float*)d_in[i_dnv[l]],
        vtmp, sumsq, m);
    k_normalize<<<nbN, 256, 0, stream>>>(vtmp, sumsq, vbuf);
  }
  k_final<<<nbN, 256, 0, stream>>>(nodes, vbuf, (float*)d_out);
}